// EDM_Block_75024488726874
// MI455X (gfx1250) — compile-verified
//
#include <hip/hip_runtime.h>

typedef __attribute__((ext_vector_type(2))) float v2f;
typedef __attribute__((ext_vector_type(4))) float fv4;
typedef __attribute__((ext_vector_type(8))) float v8f;
typedef __attribute__((ext_vector_type(4))) unsigned int u32x4;
typedef __attribute__((ext_vector_type(8))) int i32x8;
typedef __attribute__((ext_vector_type(4))) int i32x4;

#define Hdim 128

#define WMMA_F32(ACC, AV, BV) \
    ACC = __builtin_amdgcn_wmma_f32_16x16x4_f32(false, AV, false, BV, (short)0, ACC, false, false)

// TDM (tensor_load_to_lds) on the gfx1250 device pass; builtin arity differs
// between toolchains (5 args on ROCm7.2/clang-22, 6 args on clang-23+).
#if defined(__gfx1250__)
#if __has_builtin(__builtin_amdgcn_tensor_load_to_lds) && __has_builtin(__builtin_amdgcn_s_wait_tensorcnt)
#define USE_TDM 1
#endif
#endif

// ---------------------------------------------------------------------------
// Edge encoder: edge_attr[e,:] = (relu(l_e*w1 + b1) @ w2 + b2) * bond_emb[type_e]
// One wave computes a 16(edge) x 64(col) tile: the A fragment (hidden
// activations from the scalar edge length) is generated once per K-step and
// reused by 4 WMMAs. Result is stored with non-temporal hint (streamed 328MB).
// ---------------------------------------------------------------------------
__global__ __launch_bounds__(256) void edge_attr_wmma_kernel(
    const float* __restrict__ edge_length, const int* __restrict__ edge_type,
    const float* __restrict__ w1, const float* __restrict__ b1,
    const float* __restrict__ w2, const float* __restrict__ b2,
    const float* __restrict__ bond_emb, float* __restrict__ edge_attr,
    int E)
{
    const int wave       = (int)((blockIdx.x * blockDim.x + threadIdx.x) >> 5);
    const int lane       = threadIdx.x & 31;
    const int col_groups = Hdim / 64;               // 2
    const int row_tile   = wave / col_groups;
    const int cg         = wave % col_groups;
    if (row_tile * 16 >= E) return;                 // wave-uniform (EXEC stays all-1)

    const int e0   = row_tile * 16;
    const int n0   = cg * 64;
    const int m    = lane & 15;
    const int half = lane >> 4;

    const float l = edge_length[e0 + m];

    v8f acc0 = {}, acc1 = {}, acc2 = {}, acc3 = {};
    for (int k0 = 0; k0 < Hdim; k0 += 4) {
        const int ka = k0 + 2 * half;               // lanes 0-15: K=k0,k0+1; 16-31: k0+2,k0+3
        v2f a;
        a.x = fmaxf(fmaf(l, w1[ka],     b1[ka]),     0.0f);
        a.y = fmaxf(fmaf(l, w1[ka + 1], b1[ka + 1]), 0.0f);
        const float* r0 = w2 + ka * Hdim + n0 + m;
        const float* r1 = r0 + Hdim;
        v2f b0v, b1v, b2v, b3v;
        b0v.x = r0[0];  b0v.y = r1[0];
        b1v.x = r0[16]; b1v.y = r1[16];
        b2v.x = r0[32]; b2v.y = r1[32];
        b3v.x = r0[48]; b3v.y = r1[48];
        WMMA_F32(acc0, a, b0v);
        WMMA_F32(acc1, a, b1v);
        WMMA_F32(acc2, a, b2v);
        WMMA_F32(acc3, a, b3v);
    }

    const float bias0 = b2[n0 + m];
    const float bias1 = b2[n0 + m + 16];
    const float bias2 = b2[n0 + m + 32];
    const float bias3 = b2[n0 + m + 48];

    #pragma unroll
    for (int v = 0; v < 8; ++v) {                   // D row = v (+8 for upper lane half)
        const int e = e0 + v + 8 * half;
        const float* bem = bond_emb + (size_t)edge_type[e] * Hdim + n0 + m;
        float* outp = edge_attr + (size_t)e * Hdim + n0 + m;
        __builtin_nontemporal_store((acc0[v] + bias0) * bem[0],  outp + 0);
        __builtin_nontemporal_store((acc1[v] + bias1) * bem[16], outp + 16);
        __builtin_nontemporal_store((acc2[v] + bias2) * bem[32], outp + 32);
        __builtin_nontemporal_store((acc3[v] + bias3) * bem[48], outp + 48);
    }
}

// ---------------------------------------------------------------------------
// C[M x Ncol] = A[M x K] @ W[K x Ncol] + bias (optional relu).
// The 64-row A panel (64*K fp32, contiguous in memory) is staged into LDS by
// the Tensor Data Mover: wave 0 issues one TENSOR_LOAD_TO_LDS with a D#
// describing a 2-D tile {tile_dim0=K, tile_dim1=64, stride=K}, waits on
// TENSORcnt, and the workgroup barrier releases the other waves. An empty
// asm with a memory clobber makes the LDS buffer visible as TDM-written so
// the compiler cannot fold the subsequent LDS reads. Each wave then feeds
// its A fragment from LDS into 4 WMMA accumulators (16x64 tile).
// Requires M % 64 == 0, K % 4 == 0, K <= 128, Ncol in {64,128}.
//   Ncol==128: blockDim=256 (8 waves: 4 row-tiles x 2 col-groups)
//   Ncol==64 : blockDim=128 (4 waves: 4 row-tiles x 1 col-group)
// ---------------------------------------------------------------------------
__global__ __launch_bounds__(256) void gemm_bias_wmma_kernel(
    const float* __restrict__ A, const float* __restrict__ W,
    const float* __restrict__ bias, float* __restrict__ C,
    int M, int K, int Ncol, int relu)
{
    __shared__ float As[64 * 128];                  // 32 KB A panel (LDS offset 0)

    const int tid   = threadIdx.x;
    const int r_blk = blockIdx.x * 64;

#ifdef USE_TDM
    if (tid < 32) {                                 // wave 0 drives the TDM
        const unsigned long long ga =
            (unsigned long long)(uintptr_t)(A + (size_t)r_blk * K);
        // D# group0: count=1 | lds_addr=0 | global_addr[56:0] | type=2
        u32x4 g0 = { 1u,                                 // count=1, user mode
                     0u,                                 // lds_addr = 0 (only LDS alloc)
                     (unsigned int)(ga & 0xFFFFFFFFu),
                     (unsigned int)((ga >> 32) & 0x01FFFFFFu) | (2u << 30) };
        // D# group1: data_size=4B; tensor_dim0=K, tensor_dim1=M;
        //            tile_dim0=K, tile_dim1=64; tensor_dim0_stride=K
        i32x8 g1 = { (int)0x00020000u,                          // data_size=2 (4B)
                     (int)(((unsigned)K & 0xFFFFu) << 16),      // tensor_dim0 lo
                     (int)((((unsigned)K >> 16) & 0xFFFFu) |
                           (((unsigned)M & 0xFFFFu) << 16)),    // dim0 hi | dim1 lo
                     (int)((((unsigned)M >> 16) & 0xFFFFu) |
                           (((unsigned)K & 0xFFFFu) << 16)),    // dim1 hi | tile_dim0
                     64,                                        // tile_dim1=64, tile_dim2=0
                     K,                                         // dim0_stride lo32
                     0, 0 };
        i32x4 z4 = { 0, 0, 0, 0 };                              // 2-D tile: groups 2/3 unused
#if defined(__clang_major__) && (__clang_major__ >= 23)
        i32x8 z8 = { 0, 0, 0, 0, 0, 0, 0, 0 };
        __builtin_amdgcn_tensor_load_to_lds(g0, g1, z4, z4, z8, 0);
#else
        __builtin_amdgcn_tensor_load_to_lds(g0, g1, z4, z4, 0);
#endif
        __builtin_amdgcn_s_wait_tensorcnt(0);
    }
    // The TDM writes LDS through the descriptor, not through a pointer the
    // compiler can see: escape As + memory clobber so the ds_loads below
    // cannot be folded away as reads of never-written memory.
    asm volatile("" : : "r"(&As[0]) : "memory");
#else
    // cooperative, fully-coalesced float4 panel load
    {
        const int total4 = (64 * K) >> 2;
        const fv4* src4  = (const fv4*)(A + (size_t)r_blk * K);
        fv4* dst4        = (fv4*)As;
        for (int idx = tid; idx < total4; idx += blockDim.x)
            dst4[idx] = src4[idx];
    }
#endif
    __syncthreads();

    const int wave       = tid >> 5;
    const int lane       = tid & 31;
    const int col_groups = Ncol >> 6;               // 2 or 1
    const int lrow       = (col_groups == 2) ? (wave >> 1) : wave;   // 0..3
    const int cg         = (col_groups == 2) ? (wave & 1) : 0;

    const int m    = lane & 15;
    const int half = lane >> 4;
    const int n0   = cg * 64;

    const float* Arow = As + (lrow * 16 + m) * K;   // LDS

    v8f acc0 = {}, acc1 = {}, acc2 = {}, acc3 = {};
    for (int k0 = 0; k0 < K; k0 += 4) {
        const int ka = k0 + 2 * half;
        v2f a;
        a.x = Arow[ka];                             // ds_load pair from LDS
        a.y = Arow[ka + 1];
        const float* r0 = W + ka * Ncol + n0 + m;
        const float* r1 = r0 + Ncol;
        v2f b0v, b1v, b2v, b3v;
        b0v.x = r0[0];  b0v.y = r1[0];
        b1v.x = r0[16]; b1v.y = r1[16];
        b2v.x = r0[32]; b2v.y = r1[32];
        b3v.x = r0[48]; b3v.y = r1[48];
        WMMA_F32(acc0, a, b0v);
        WMMA_F32(acc1, a, b1v);
        WMMA_F32(acc2, a, b2v);
        WMMA_F32(acc3, a, b3v);
    }

    const float bias0 = bias[n0 + m];
    const float bias1 = bias[n0 + m + 16];
    const float bias2 = bias[n0 + m + 32];
    const float bias3 = bias[n0 + m + 48];

    #pragma unroll
    for (int v = 0; v < 8; ++v) {
        const int r = r_blk + lrow * 16 + v + 8 * half;
        float* outp = C + (size_t)r * Ncol + n0 + m;
        float v0 = acc0[v] + bias0;
        float v1 = acc1[v] + bias1;
        float v2 = acc2[v] + bias2;
        float v3 = acc3[v] + bias3;
        if (relu) {
            v0 = fmaxf(v0, 0.0f); v1 = fmaxf(v1, 0.0f);
            v2 = fmaxf(v2, 0.0f); v3 = fmaxf(v3, 0.0f);
        }
        outp[0]  = v0;
        outp[16] = v1;
        outp[32] = v2;
        outp[48] = v3;
    }
}

// --------------------------- small helper kernels ---------------------------
__global__ void node_embed_kernel(const float* __restrict__ ne,
                                  const float* __restrict__ w,
                                  const float* __restrict__ b,
                                  float* __restrict__ h, int total)
{
    int i = blockIdx.x * blockDim.x + threadIdx.x;
    if (i >= total) return;
    int node = i >> 7, c = i & (Hdim - 1);
    float acc = b[c];
    #pragma unroll
    for (int k = 0; k < 8; ++k)
        acc = fmaf(ne[node * 8 + k], w[k * Hdim + c], acc);
    h[i] = acc;
}

__global__ void zero_kernel(float* __restrict__ p, int total)
{
    int i = blockIdx.x * blockDim.x + threadIdx.x;
    if (i < total) p[i] = 0.0f;
}

// msg = hw[src] + edge_attr ; agg[dst] += msg.
// edge_attr is streamed once per conv -> non-temporal load (keep hw/agg in L2);
// hw gather + agg atomics stay cached (both L2-resident, 20 MB each).
__global__ void scatter_kernel(const float* __restrict__ hw,
                               const float* __restrict__ edge_attr,
                               const int* __restrict__ src,
                               const int* __restrict__ dst,
                               float* __restrict__ agg, int E)
{
    int tid = blockIdx.x * blockDim.x + threadIdx.x;
    if (tid >= E * (Hdim / 4)) return;
    const int e = tid >> 5;                         // H/4 = 32 chunks per edge
    const int c = (tid & 31) << 2;
    const int s = src[e], d = dst[e];
    const fv4 hv = *(const fv4*)(hw + (size_t)s * Hdim + c);
    const fv4 ev = __builtin_nontemporal_load(
        (const fv4*)(edge_attr + (size_t)e * Hdim + c));
    float* out = agg + (size_t)d * Hdim + c;
    unsafeAtomicAdd(out + 0, hv.x + ev.x);
    unsafeAtomicAdd(out + 1, hv.y + ev.y);
    unsafeAtomicAdd(out + 2, hv.z + ev.z);
    unsafeAtomicAdd(out + 3, hv.w + ev.w);
}

__global__ void residual_kernel(float* __restrict__ h,
                                const float* __restrict__ agg, int total)
{
    int i = blockIdx.x * blockDim.x + threadIdx.x;
    if (i < total) h[i] = fmaxf(agg[i], 0.0f) + h[i];
}

// delta = x2 @ head_w3 + b3 ; out = pos + delta
__global__ void head3_pos_kernel(const float* __restrict__ x2,
                                 const float* __restrict__ w3,
                                 const float* __restrict__ b3,
                                 const float* __restrict__ pos,
                                 float* __restrict__ out, int n)
{
    int i = blockIdx.x * blockDim.x + threadIdx.x;
    if (i >= n) return;
    const float* xr = x2 + (size_t)i * 64;
    float a0 = b3[0], a1 = b3[1];
    #pragma unroll 8
    for (int k = 0; k < 64; ++k) {
        float xv = xr[k];
        a0 = fmaf(xv, w3[2 * k], a0);
        a1 = fmaf(xv, w3[2 * k + 1], a1);
    }
    out[2 * i]     = pos[2 * i]     + a0;
    out[2 * i + 1] = pos[2 * i + 1] + a1;
}

// ---------------------------------------------------------------------------
extern "C" void kernel_launch(void* const* d_in, const int* in_sizes, int n_in,
                              void* d_out, int out_size, void* d_ws, size_t ws_size,
                              hipStream_t stream)
{
    const float* node_emb    = (const float*)d_in[0];
    const float* pos         = (const float*)d_in[3];
    const int*   edge_index  = (const int*)  d_in[4];
    const int*   edge_type   = (const int*)  d_in[5];
    const float* edge_length = (const float*)d_in[6];
    const float* bond_emb    = (const float*)d_in[9];
    const float* edge_w1     = (const float*)d_in[10];
    const float* edge_b1     = (const float*)d_in[11];
    const float* edge_w2     = (const float*)d_in[12];
    const float* edge_b2     = (const float*)d_in[13];
    const float* node_w      = (const float*)d_in[14];
    const float* node_b      = (const float*)d_in[15];
    const float* conv_w      = (const float*)d_in[16];
    const float* conv_b      = (const float*)d_in[17];
    const float* head_w1     = (const float*)d_in[18];
    const float* head_b1     = (const float*)d_in[19];
    const float* head_w2     = (const float*)d_in[20];
    const float* head_b2     = (const float*)d_in[21];
    const float* head_w3     = (const float*)d_in[22];
    const float* head_b3     = (const float*)d_in[23];

    const int N = in_sizes[1];      // 40000
    const int E = in_sizes[5];      // 640000
    const int* src = edge_index;
    const int* dst = edge_index + E;

    // workspace layout (fp32): edge_attr | h | hw | agg   (x1 reuses hw, x2 reuses agg)
    float* edge_attr = (float*)d_ws;
    float* h   = edge_attr + (size_t)E * Hdim;
    float* hw  = h   + (size_t)N * Hdim;
    float* agg = hw  + (size_t)N * Hdim;
    float* x1  = hw;
    float* x2  = agg;

    const int NH = N * Hdim;

    // 1) edge encoder GEMM + bond gating  (one wave per 16x64 tile)
    {
        int waves  = (E / 16) * (Hdim / 64);        // 80000
        int blocks = (waves + 7) / 8;
        edge_attr_wmma_kernel<<<blocks, 256, 0, stream>>>(
            edge_length, edge_type, edge_w1, edge_b1, edge_w2, edge_b2,
            bond_emb, edge_attr, E);
    }

    // 2) node embedding h = node_emb @ node_w + node_b
    node_embed_kernel<<<(NH + 255) / 256, 256, 0, stream>>>(
        node_emb, node_w, node_b, h, NH);

    // 3) 4 conv layers
    for (int i = 0; i < 4; ++i) {
        gemm_bias_wmma_kernel<<<N / 64, 256, 0, stream>>>(
            h, conv_w + (size_t)i * Hdim * Hdim, conv_b + (size_t)i * Hdim,
            hw, N, Hdim, Hdim, 0);

        zero_kernel<<<(NH + 255) / 256, 256, 0, stream>>>(agg, NH);

        int sthreads = E * (Hdim / 4);
        scatter_kernel<<<(sthreads + 255) / 256, 256, 0, stream>>>(
            hw, edge_attr, src, dst, agg, E);

        residual_kernel<<<(NH + 255) / 256, 256, 0, stream>>>(h, agg, NH);
    }

    // 4) head MLP
    {
        gemm_bias_wmma_kernel<<<N / 64, 256, 0, stream>>>(
            h, head_w1, head_b1, x1, N, Hdim, Hdim, 1);

        gemm_bias_wmma_kernel<<<N / 64, 128, 0, stream>>>(
            x1, head_w2, head_b2, x2, N, Hdim, 64, 1);

        head3_pos_kernel<<<(N + 255) / 256, 256, 0, stream>>>(
            x2, head_w3, head_b3, pos, (float*)d_out, N);
    }
}